// DiscriminativeLoss_65884798320741
// MI455X (gfx1250) — compile-verified
//
#include <hip/hip_runtime.h>

#define DELTA_V 0.5f
#define DELTA_D 3.0f
#define PRM_VAR 1.0f
#define PRM_DIST 1.0f
#define PRM_REG 0.001f

typedef __attribute__((ext_vector_type(2))) float v2f;
typedef __attribute__((ext_vector_type(8))) float v8f;

// ---------------- workspace layout (float offsets) ----------------
enum {
  WS_BIN_SUM   = 0,                    // [4][2][3]  (d0,d1 sums, count)
  WS_BIN_VAR   = WS_BIN_SUM + 24,      // [4][2]
  WS_BIN_MEAN  = WS_BIN_VAR + 8,       // [4][2][2]
  WS_BIN_BASE  = WS_BIN_MEAN + 16,     // [4]   (dist+reg terms)
  WS_INST_SUM  = WS_BIN_BASE + 4,      // [4][6][6]  (5 sums, count)
  WS_INST_VAR  = WS_INST_SUM + 144,    // [4][6]
  WS_INST_MEAN = WS_INST_VAR + 24,     // [4][6][5]
  WS_INST_BASE = WS_INST_MEAN + 120,   // [4]
  WS_TOTAL     = WS_INST_BASE + 4
};

// ---------------- WMMA block reduction (two-stage) ----------------
template <int P, int NT>
struct RedLDS {
  static constexpr int NTILE = (P + 15) / 16;
  static constexpr int COLS  = NTILE * 16;
  static constexpr int PPAD  = COLS + 1;
  static constexpr int WAVES = NT / 32;
  static constexpr int SIZE  = NT * PPAD + WAVES * PPAD;
};

// Per-thread partials acc[P] -> one atomicAdd per column into gout[P].
// Stage 1: each wave32 reduces its own 32 rows via chained
// v_wmma_f32_16x16x4_f32 with A = ones (D[m][n] = sum_k B[k][n] + C[m][n],
// invariant to the exact row->(VGPR,half) mapping; only column = lane%16
// matters). Stage 2: wave 0 reduces the WAVES per-wave rows.
template <int P, int NT>
__device__ inline void block_reduce_add(float* smem, const float (&acc)[P],
                                        float* __restrict__ gout) {
  using L = RedLDS<P, NT>;
  static_assert(L::WAVES % 4 == 0, "need >=4 waves for stage 2");
  const int tid  = threadIdx.x;
  const int wave = tid >> 5;
  const int lane = tid & 31;
  const int col  = lane & 15;
  const int half = lane >> 4;

#pragma unroll
  for (int p = 0; p < P; ++p) smem[tid * L::PPAD + p] = acc[p];
#pragma unroll
  for (int p = P; p < L::COLS; ++p) smem[tid * L::PPAD + p] = 0.f;  // pad cols
  __syncthreads();

  float* wrow = smem + NT * L::PPAD;  // [WAVES][PPAD] per-wave column sums
  v2f a; a[0] = 1.0f; a[1] = 1.0f;    // A = ones(16x4)

#pragma unroll
  for (int tile = 0; tile < L::NTILE; ++tile) {
    v8f c = {0.f, 0.f, 0.f, 0.f, 0.f, 0.f, 0.f, 0.f};
#pragma unroll
    for (int i = 0; i < 8; ++i) {     // 4 rows (threads) per WMMA
      const int r = wave * 32 + i * 4 + half * 2;
      v2f bv;
      bv[0] = smem[r * L::PPAD + tile * 16 + col];
      bv[1] = smem[(r + 1) * L::PPAD + tile * 16 + col];
      c = __builtin_amdgcn_wmma_f32_16x16x4_f32(false, a, false, bv,
                                                (short)0, c, false, false);
    }
    if (lane < 16) wrow[wave * L::PPAD + tile * 16 + lane] = c[0];
  }
  __syncthreads();

  if (wave == 0) {
#pragma unroll
    for (int tile = 0; tile < L::NTILE; ++tile) {
      v8f c = {0.f, 0.f, 0.f, 0.f, 0.f, 0.f, 0.f, 0.f};
#pragma unroll
      for (int i = 0; i < L::WAVES / 4; ++i) {
        const int r = i * 4 + half * 2;
        v2f bv;
        bv[0] = wrow[r * L::PPAD + tile * 16 + col];
        bv[1] = wrow[(r + 1) * L::PPAD + tile * 16 + col];
        c = __builtin_amdgcn_wmma_f32_16x16x4_f32(false, a, false, bv,
                                                  (short)0, c, false, false);
      }
      const int p = tile * 16 + col;
      if (lane < 16 && p < P) atomicAdd(&gout[p], c[0]);
    }
  }
}

// ---------------- pass 1: segment sums + counts ----------------
template <int K, int D, int NT>
__global__ void pass1_kernel(const float* __restrict__ pred,
                             const int* __restrict__ labels,
                             float* __restrict__ out_sum,  // [B][K][D+1]
                             int N, int blocksPerBatch) {
  constexpr int P = K * (D + 1);
  __shared__ float smem[RedLDS<P, NT>::SIZE];
  const int b  = blockIdx.x / blocksPerBatch;
  const int bb = blockIdx.x % blocksPerBatch;
  const float* pb = pred + (size_t)b * D * N;
  const int*   lb = labels + (size_t)b * N;

  float acc[P];
#pragma unroll
  for (int p = 0; p < P; ++p) acc[p] = 0.f;

  const int Nvec   = N & ~3;
  const int stride = blocksPerBatch * NT * 4;
  for (int n = (bb * NT + threadIdx.x) * 4; n < Nvec; n += stride) {
    int ll[4];
    *(int4*)ll = *(const int4*)(lb + n);                 // global_load_b128
    float vv[D][4];
#pragma unroll
    for (int d = 0; d < D; ++d)
      *(float4*)vv[d] = *(const float4*)(pb + (size_t)d * N + n);
#pragma unroll
    for (int j = 0; j < 4; ++j) {
#pragma unroll
      for (int k = 0; k < K; ++k) {
        const float m = (ll[j] == k) ? 1.f : 0.f;
#pragma unroll
        for (int d = 0; d < D; ++d) acc[k * (D + 1) + d] += m * vv[d][j];
        acc[k * (D + 1) + D] += m;
      }
    }
  }
  if (bb == 0) {  // scalar tail (N % 4), handled by batch's block 0
    for (int n = Nvec + threadIdx.x; n < N; n += NT) {
      const int lab = lb[n];
#pragma unroll
      for (int k = 0; k < K; ++k) {
        const float m = (lab == k) ? 1.f : 0.f;
#pragma unroll
        for (int d = 0; d < D; ++d)
          acc[k * (D + 1) + d] += m * pb[(size_t)d * N + n];
        acc[k * (D + 1) + D] += m;
      }
    }
  }
  block_reduce_add<P, NT>(smem, acc, out_sum + (size_t)b * P);
}

// ---------------- means + push(dist) + reg terms (tiny) ----------------
template <int K, int D>
__device__ inline void means_base(const float* __restrict__ sum,  // [K][D+1]
                                  float* __restrict__ mean,       // [K][D]
                                  float* __restrict__ base) {
  float mu[K][D];
#pragma unroll
  for (int k = 0; k < K; ++k) {
    const float cnt = sum[k * (D + 1) + D];
#pragma unroll
    for (int d = 0; d < D; ++d) {
      mu[k][d] = sum[k * (D + 1) + d] / cnt;
      mean[k * D + d] = mu[k][d];
    }
  }
  float ld = 0.f;
#pragma unroll
  for (int k = 0; k < K; ++k)
#pragma unroll
    for (int j = 0; j < K; ++j) {
      if (j == k) continue;
      float sq = 0.f;
#pragma unroll
      for (int d = 0; d < D; ++d) {
        const float df = mu[k][d] - mu[j][d];
        sq += df * df;
      }
      const float dn = fmaxf(2.f * DELTA_D - sqrtf(sq), 0.f);
      ld += dn * dn;
    }
  ld /= (float)(K * (K - 1));
  float lr = 0.f;
#pragma unroll
  for (int k = 0; k < K; ++k) {
    float sq = 0.f;
#pragma unroll
    for (int d = 0; d < D; ++d) sq += mu[k][d] * mu[k][d];
    lr += sqrtf(sq);
  }
  lr /= (float)K;
  *base = PRM_DIST * ld + PRM_REG * lr;
}

__global__ void means_kernel(float* __restrict__ ws) {
  const int t = threadIdx.x;
  if (t < 4) {
    means_base<2, 2>(ws + WS_BIN_SUM + t * 6, ws + WS_BIN_MEAN + t * 4,
                     ws + WS_BIN_BASE + t);
  } else if (t < 8) {
    const int b = t - 4;
    means_base<6, 5>(ws + WS_INST_SUM + b * 36, ws + WS_INST_MEAN + b * 30,
                     ws + WS_INST_BASE + b);
  }
}

// ---------------- pass 2: hinged variance segment sums ----------------
template <int K, int D, int NT>
__global__ void pass2_kernel(const float* __restrict__ pred,
                             const int* __restrict__ labels,
                             const float* __restrict__ mean_ws,  // [B][K][D]
                             float* __restrict__ var_out,        // [B][K]
                             int N, int blocksPerBatch) {
  constexpr int P = K;
  __shared__ float smem[RedLDS<P, NT>::SIZE];
  __shared__ float smean[K * D];
  const int b  = blockIdx.x / blocksPerBatch;
  const int bb = blockIdx.x % blocksPerBatch;
  if (threadIdx.x < K * D)
    smean[threadIdx.x] = mean_ws[(size_t)b * K * D + threadIdx.x];
  __syncthreads();

  const float* pb = pred + (size_t)b * D * N;
  const int*   lb = labels + (size_t)b * N;

  float acc[P];
#pragma unroll
  for (int p = 0; p < P; ++p) acc[p] = 0.f;

  const int Nvec   = N & ~3;
  const int stride = blocksPerBatch * NT * 4;
  for (int n = (bb * NT + threadIdx.x) * 4; n < Nvec; n += stride) {
    int ll[4];
    *(int4*)ll = *(const int4*)(lb + n);
    float vv[D][4];
#pragma unroll
    for (int d = 0; d < D; ++d)
      *(float4*)vv[d] = *(const float4*)(pb + (size_t)d * N + n);
#pragma unroll
    for (int j = 0; j < 4; ++j) {
      const int lab = ll[j];
      float sq = 0.f;
#pragma unroll
      for (int d = 0; d < D; ++d) {
        const float df = vv[d][j] - smean[lab * D + d];
        sq += df * df;
      }
      const float h  = fmaxf(sqrtf(sq) - DELTA_V, 0.f);
      const float h2 = h * h;
#pragma unroll
      for (int k = 0; k < K; ++k) acc[k] += (lab == k) ? h2 : 0.f;
    }
  }
  if (bb == 0) {
    for (int n = Nvec + threadIdx.x; n < N; n += NT) {
      const int lab = lb[n];
      float sq = 0.f;
#pragma unroll
      for (int d = 0; d < D; ++d) {
        const float df = pb[(size_t)d * N + n] - smean[lab * D + d];
        sq += df * df;
      }
      const float h  = fmaxf(sqrtf(sq) - DELTA_V, 0.f);
      const float h2 = h * h;
#pragma unroll
      for (int k = 0; k < K; ++k) acc[k] += (lab == k) ? h2 : 0.f;
    }
  }
  block_reduce_add<P, NT>(smem, acc, var_out + (size_t)b * P);
}

// ---------------- zero + finalize ----------------
__global__ void zero_ws_kernel(float* __restrict__ ws) {
  for (int i = threadIdx.x; i < WS_TOTAL; i += blockDim.x) ws[i] = 0.f;
}

__global__ void finalize_kernel(const float* __restrict__ ws,
                                float* __restrict__ out) {
  if (threadIdx.x == 0) {
    float bl = 0.f, il = 0.f;
    for (int b = 0; b < 4; ++b) {
      float lv = 0.f;
      for (int k = 0; k < 2; ++k) {
        const float cnt = ws[WS_BIN_SUM + b * 6 + k * 3 + 2];
        lv += ws[WS_BIN_VAR + b * 2 + k] / cnt;
      }
      bl += PRM_VAR * (lv * 0.5f) + ws[WS_BIN_BASE + b];
      float lv2 = 0.f;
      for (int k = 0; k < 6; ++k) {
        const float cnt = ws[WS_INST_SUM + b * 36 + k * 6 + 5];
        lv2 += ws[WS_INST_VAR + b * 6 + k] / cnt;
      }
      il += PRM_VAR * (lv2 / 6.f) + ws[WS_INST_BASE + b];
    }
    out[0] = bl * 0.25f;
    out[1] = il * 0.25f;
  }
}

// ---------------- launch ----------------
extern "C" void kernel_launch(void* const* d_in, const int* in_sizes, int n_in,
                              void* d_out, int out_size, void* d_ws, size_t ws_size,
                              hipStream_t stream) {
  const float* bin_logits  = (const float*)d_in[0];
  const int*   bin_labels  = (const int*)d_in[1];
  const float* inst_logits = (const float*)d_in[2];
  const int*   inst_labels = (const int*)d_in[3];
  float* out = (float*)d_out;
  float* ws  = (float*)d_ws;

  const int B = 4;
  const int N = in_sizes[1] / B;  // binary_labels is [B, N]
  constexpr int NT  = 256;        // 8 wave32s per block
  constexpr int BPB = 256;        // blocks per batch

  zero_ws_kernel<<<1, 256, 0, stream>>>(ws);
  pass1_kernel<2, 2, NT><<<B * BPB, NT, 0, stream>>>(
      bin_logits, bin_labels, ws + WS_BIN_SUM, N, BPB);
  pass1_kernel<6, 5, NT><<<B * BPB, NT, 0, stream>>>(
      inst_logits, inst_labels, ws + WS_INST_SUM, N, BPB);
  means_kernel<<<1, 8, 0, stream>>>(ws);
  pass2_kernel<2, 2, NT><<<B * BPB, NT, 0, stream>>>(
      bin_logits, bin_labels, ws + WS_BIN_MEAN, ws + WS_BIN_VAR, N, BPB);
  pass2_kernel<6, 5, NT><<<B * BPB, NT, 0, stream>>>(
      inst_logits, inst_labels, ws + WS_INST_MEAN, ws + WS_INST_VAR, N, BPB);
  finalize_kernel<<<1, 1, 0, stream>>>(ws, out);
}